// VegasMap_7524782702847
// MI455X (gfx1250) — compile-verified
//
#include <hip/hip_runtime.h>

#define BLOCK 256
#define ROWS_PER_THREAD 8

typedef __attribute__((ext_vector_type(4))) float f32x4;
typedef int i32x4 __attribute__((vector_size(16)));

#if defined(__has_builtin)
#if __has_builtin(__builtin_amdgcn_global_load_async_to_lds_b32) && \
    __has_builtin(__builtin_amdgcn_s_wait_asynccnt)
#define USE_ASYNC_LDS 1
#endif
#if __has_builtin(__builtin_amdgcn_global_load_async_to_lds_b128)
#define USE_ASYNC_LDS_B128 1
#endif
#endif

typedef __attribute__((address_space(1))) int glb_i32;
typedef __attribute__((address_space(3))) int lds_i32;
typedef __attribute__((address_space(1))) i32x4 glb_i32x4;
typedef __attribute__((address_space(3))) i32x4 lds_i32x4;

// Stage grid table (D*(ninc+1) floats) from global into LDS via the CDNA5
// async-to-LDS path (ASYNCcnt), b128-wide where available.
__device__ __forceinline__ void stage_grid_to_lds(const float* __restrict__ grid,
                                                  float* sgrid, int tsz) {
#ifdef USE_ASYNC_LDS
#ifdef USE_ASYNC_LDS_B128
    const int nvec = tsz >> 2;  // float4 chunks (16-byte aligned: LDS base + i*16)
    for (int i = threadIdx.x; i < nvec; i += BLOCK) {
        __builtin_amdgcn_global_load_async_to_lds_b128(
            (glb_i32x4*)(grid + 4 * i),
            (lds_i32x4*)(sgrid + 4 * i),
            /*offset=*/0, /*cpol=*/0);
    }
    for (int i = (nvec << 2) + threadIdx.x; i < tsz; i += BLOCK) {
        __builtin_amdgcn_global_load_async_to_lds_b32(
            (glb_i32*)(grid + i),
            (lds_i32*)(sgrid + i), 0, 0);
    }
#else
    for (int i = threadIdx.x; i < tsz; i += BLOCK) {
        __builtin_amdgcn_global_load_async_to_lds_b32(
            (glb_i32*)(grid + i),
            (lds_i32*)(sgrid + i), 0, 0);
    }
#endif
    __builtin_amdgcn_s_wait_asynccnt(0);
#else
    for (int i = threadIdx.x; i < tsz; i += BLOCK) sgrid[i] = grid[i];
#endif
    __syncthreads();
}

// Fast path: compile-time D (multiple of 4) -> float4 NT streams, fully unrolled.
template <int D>
__global__ __launch_bounds__(BLOCK) void vegas_map_kernel(
    const float* __restrict__ y, const float* __restrict__ grid,
    float* __restrict__ x, float* __restrict__ jac,
    int ninc, long long nrows) {
    extern __shared__ float sgrid[];  // [D][ninc+1]
    const int tw = ninc + 1;
    stage_grid_to_lds(grid, sgrid, D * tw);

    const float nincf = (float)ninc;
    const long long stride = (long long)gridDim.x * BLOCK;
    for (long long row = (long long)blockIdx.x * BLOCK + threadIdx.x;
         row < nrows; row += stride) {
        // Prefetch next grid-stride row of y (global_prefetch_b8).
        if (row + stride < nrows)
            __builtin_prefetch(y + (row + stride) * D, 0, 3);

        const f32x4* __restrict__ yrow = (const f32x4*)(y + row * (long long)D);
        f32x4 yv[D / 4];
#pragma unroll
        for (int q = 0; q < D / 4; ++q)
            yv[q] = __builtin_nontemporal_load(yrow + q);  // streamed once: NT

        f32x4 xv[D / 4];
        float jv = 1.0f;
#pragma unroll
        for (int d = 0; d < D; ++d) {
            float t = yv[d >> 2][d & 3] * nincf;       // y * ninc
            int iy = (int)floorf(t);
            float dy = t - (float)iy;                  // fractional part
            int iyc = iy < ninc ? iy : (ninc - 1);
            if (iyc < 0) iyc = 0;
            const float* gd = sgrid + d * tw;
            float g0 = gd[iyc];                        // ds_load (2addr pair)
            float g1 = gd[iyc + 1];
            float ig = g1 - g0;                        // == inc[d][iyc] bit-exactly
            // in-range: g0 + ig*dy ; iy==ninc: x = grid[d][ninc] == g1 exactly
            float xr = (iy < ninc) ? fmaf(ig, dy, g0) : g1;
            xv[d >> 2][d & 3] = xr;
            jv *= ig * nincf;                          // fac identical in both branches
        }

        f32x4* __restrict__ xrow = (f32x4*)(x + row * (long long)D);
#pragma unroll
        for (int q = 0; q < D / 4; ++q)
            __builtin_nontemporal_store(xv[q], xrow + q);
        __builtin_nontemporal_store(jv, jac + row);
    }
}

// Generic fallback for arbitrary runtime D.
__global__ __launch_bounds__(BLOCK) void vegas_map_kernel_gen(
    const float* __restrict__ y, const float* __restrict__ grid,
    float* __restrict__ x, float* __restrict__ jac,
    int D, int ninc, long long nrows) {
    extern __shared__ float sgrid[];
    const int tw = ninc + 1;
    stage_grid_to_lds(grid, sgrid, D * tw);

    const float nincf = (float)ninc;
    const long long stride = (long long)gridDim.x * BLOCK;
    for (long long row = (long long)blockIdx.x * BLOCK + threadIdx.x;
         row < nrows; row += stride) {
        float jv = 1.0f;
        for (int d = 0; d < D; ++d) {
            float t = y[row * (long long)D + d] * nincf;
            int iy = (int)floorf(t);
            float dy = t - (float)iy;
            int iyc = iy < ninc ? iy : (ninc - 1);
            if (iyc < 0) iyc = 0;
            const float* gd = sgrid + d * tw;
            float g0 = gd[iyc];
            float g1 = gd[iyc + 1];
            float ig = g1 - g0;
            x[row * (long long)D + d] = (iy < ninc) ? fmaf(ig, dy, g0) : g1;
            jv *= ig * nincf;
        }
        jac[row] = jv;
    }
}

extern "C" void kernel_launch(void* const* d_in, const int* in_sizes, int n_in,
                              void* d_out, int out_size, void* d_ws, size_t ws_size,
                              hipStream_t stream) {
    const float* y    = (const float*)d_in[0];
    const float* grid = (const float*)d_in[1];
    // d_in[2] (inc) is bit-exactly the first difference of grid -> recomputed
    // in-kernel; d_in[3] (ninc) derived on host from sizes (graph-capture safe).
    int D = in_sizes[1] - in_sizes[2];            // (ninc+1)*D - ninc*D
    if (D <= 0) D = 1;
    int ninc = in_sizes[2] / D;
    long long B = (long long)in_sizes[0] / D;

    float* x   = (float*)d_out;
    float* jac = x + B * (long long)D;

    size_t lds = (size_t)D * (ninc + 1) * sizeof(float);
    long long nb = (B + (long long)BLOCK * ROWS_PER_THREAD - 1) /
                   ((long long)BLOCK * ROWS_PER_THREAD);
    if (nb < 1) nb = 1;
    if (nb > 1048576) nb = 1048576;

    if (D == 16) {
        vegas_map_kernel<16><<<dim3((unsigned)nb), dim3(BLOCK), lds, stream>>>(
            y, grid, x, jac, ninc, B);
    } else {
        vegas_map_kernel_gen<<<dim3((unsigned)nb), dim3(BLOCK), lds, stream>>>(
            y, grid, x, jac, D, ninc, B);
    }
}